// DRlocal_net_79173427135059
// MI455X (gfx1250) — compile-verified
//
#include <hip/hip_runtime.h>
#include <hip/hip_bf16.h>
#include <math.h>

typedef __attribute__((ext_vector_type(2))) float v2f;
typedef __attribute__((ext_vector_type(8))) float v8f;

#define D 128
#define LDSTRIDE 130   // even pad: keeps ds 2-addr b64 alignment, bank = (k+2r)%64

static __device__ __forceinline__ v8f wmma4(v2f a, v2f b, v8f c) {
    // V_WMMA_F32_16X16X4_F32 : D = A(16x4,f32) * B(4x16,f32) + C(16x16,f32)
    return __builtin_amdgcn_wmma_f32_16x16x4_f32(false, a, false, b, (short)0, c,
                                                 false, false);
}

// A fragment 16x4 at (row base m0, k) from LDS. Lanes 0-15: rows; K pair {kh,kh+1},
// kh = 0 (lanes<16) / 2 (lanes>=16) per ISA 32-bit A layout.
static __device__ __forceinline__ v2f loadA(const float* __restrict__ s, int m0,
                                            int k, int lane) {
    const int r  = m0 + (lane & 15);
    const int kh = (lane >> 4) << 1;
    v2f a;
    a.x = s[r * LDSTRIDE + k + kh];
    a.y = s[r * LDSTRIDE + k + kh + 1];
    return a;
}

// Packed-B fragment: one coalesced 8B load per lane (global_load_b64).
static __device__ __forceinline__ v2f loadBP(const v2f* __restrict__ p, int ktiles,
                                             int nt, int ks, int lane) {
    return p[((size_t)nt * ktiles + ks) * 32 + lane];
}

// ---- pack B (row-major [K x Nc]) into fragment-major lane layout --------------
__global__ __launch_bounds__(256) void k_pack_direct(const float* __restrict__ B,
                                                     float* __restrict__ out,
                                                     int ktiles, int ntiles, int Nc) {
    int idx = blockIdx.x * 256 + threadIdx.x;
    if (idx >= ntiles * ktiles * 32) return;
    int nt = idx / (ktiles * 32), rem = idx - nt * ktiles * 32;
    int ks = rem >> 5, lane = rem & 31;
    int k = ks * 4 + ((lane >> 4) << 1);
    int n = nt * 16 + (lane & 15);
    out[(size_t)idx * 2 + 0] = B[(size_t)k * Nc + n];
    out[(size_t)idx * 2 + 1] = B[(size_t)(k + 1) * Nc + n];
}

// ---- pack B where logical B[k][n] = w[n*ldw + k] (fuses the transpose) --------
__global__ __launch_bounds__(256) void k_pack_transposed(const float* __restrict__ w,
                                                         float* __restrict__ out,
                                                         int ktiles, int ntiles, int ldw) {
    int idx = blockIdx.x * 256 + threadIdx.x;
    if (idx >= ntiles * ktiles * 32) return;
    int nt = idx / (ktiles * 32), rem = idx - nt * ktiles * 32;
    int ks = rem >> 5, lane = rem & 31;
    int k = ks * 4 + ((lane >> 4) << 1);
    int n = nt * 16 + (lane & 15);
    out[(size_t)idx * 2 + 0] = w[(size_t)n * ldw + k];
    out[(size_t)idx * 2 + 1] = w[(size_t)n * ldw + k + 1];
}

// ---------------- K1: hW = ent_embs[node_id] @ W  (2 waves x 32 rows) ----------
__global__ __launch_bounds__(64) void k_gather_gemm(
    const float* __restrict__ ent, const int* __restrict__ node_id,
    const v2f* __restrict__ Wp, float* __restrict__ hW, int N) {
    __shared__ float sA[64 * LDSTRIDE];

    const int tid = threadIdx.x;
    const int rowbase = blockIdx.x * 64;
    const bool full = (rowbase + 64) <= N;   // uniform: interior block fast path

    // stage 64 gathered rows: 2048 float4, coalesced
    if (full) {
#pragma unroll 4
        for (int i = 0; i < 32; ++i) {
            int flat = i * 64 + tid;            // float4 index
            int r = flat >> 5, c4 = flat & 31;
            int nid = node_id[rowbase + r];
            float4 v = *(const float4*)(ent + (size_t)nid * D + c4 * 4);
            float* dst = &sA[r * LDSTRIDE + c4 * 4];
            dst[0] = v.x; dst[1] = v.y; dst[2] = v.z; dst[3] = v.w;
        }
    } else {
        for (int i = 0; i < 32; ++i) {
            int flat = i * 64 + tid;
            int r = flat >> 5, c4 = flat & 31;
            int gr = rowbase + r;
            int nid = node_id[gr < N ? gr : 0];
            float4 v = *(const float4*)(ent + (size_t)nid * D + c4 * 4);
            float* dst = &sA[r * LDSTRIDE + c4 * 4];
            dst[0] = v.x; dst[1] = v.y; dst[2] = v.z; dst[3] = v.w;
        }
    }
    __syncthreads();

    const int wave = tid >> 5, lane = tid & 31;
    const int wm = wave * 32;               // wave's 32-row tile
    const int nc = lane & 15, roff = (lane >> 4) * 8;

    for (int nt = 0; nt < 8; ++nt) {
        v8f c0 = {}, c1 = {};
#pragma unroll 8
        for (int ks = 0; ks < 32; ++ks) {
            const int k = ks * 4;
            v2f b  = loadBP(Wp, 32, nt, ks, lane);   // one b64, 256B/wave, L1-hot
            v2f a0 = loadA(sA, wm, k, lane);
            v2f a1 = loadA(sA, wm + 16, k, lane);
            c0 = wmma4(a0, b, c0);
            c1 = wmma4(a1, b, c1);
        }
        if (full) {
#pragma unroll
            for (int v = 0; v < 8; ++v) {
                int gr0 = rowbase + wm + v + roff;
                hW[(size_t)gr0 * D + nt * 16 + nc] = c0[v];
                hW[(size_t)(gr0 + 16) * D + nt * 16 + nc] = c1[v];
            }
        } else {
#pragma unroll
            for (int v = 0; v < 8; ++v) {
                int gr0 = rowbase + wm + v + roff;
                int gr1 = gr0 + 16;
                if (gr0 < N) hW[(size_t)gr0 * D + nt * 16 + nc] = c0[v];
                if (gr1 < N) hW[(size_t)gr1 * D + nt * 16 + nc] = c1[v];
            }
        }
    }
}

// ---------------- K2: agg[dst] += hW[src] per edge (1 wave / edge) -------------
__global__ __launch_bounds__(256) void k_scatter(
    const float* __restrict__ hW, const int* __restrict__ esrc,
    const int* __restrict__ edst, float* __restrict__ agg, int E) {
    int idx = blockIdx.x * 256 + threadIdx.x;
    int e = idx >> 5, lane = idx & 31;
    if (e >= E) return;
    int s = esrc[e], d = edst[e];
    const float4 m = *(const float4*)(hW + (size_t)s * D + lane * 4);
    float* p = agg + (size_t)d * D + lane * 4;
    unsafeAtomicAdd(p + 0, m.x);
    unsafeAtomicAdd(p + 1, m.y);
    unsafeAtomicAdd(p + 2, m.z);
    unsafeAtomicAdd(p + 3, m.w);
}

// ---------------- K3: fused GRU cell + relu (1 wave x 32 rows / block) ---------
__global__ __launch_bounds__(32) void k_gru(
    const float* __restrict__ agg, const float* __restrict__ out_norm,
    const float* __restrict__ e_r_bias,
    const v2f* __restrict__ ihp,   // packed w_ih^T fragments: 24 ntiles x 64 ktiles
    const v2f* __restrict__ hhp,   // packed w_hh^T fragments: 24 ntiles x 32 ktiles
    const float* __restrict__ b_ih, const float* __restrict__ b_hh,
    float* __restrict__ out, int N) {
    __shared__ float sAgg[32 * LDSTRIDE];
    __shared__ float sBias[32 * LDSTRIDE];

    const int lane = threadIdx.x;
    const int rowbase = blockIdx.x * 32;
    const bool full = (rowbase + 32) <= N;   // uniform fast path

    // stage agg*out_norm and e_r_bias: 1024 float4 each, coalesced
    for (int i = 0; i < 32; ++i) {
        int flat = i * 32 + lane;           // float4 index within 32x128 tile
        int r = flat >> 5, c4 = flat & 31;
        int gr = rowbase + r;
        int cr = full ? gr : (gr < N ? gr : 0);
        float on = out_norm[cr];
        float4 va = *(const float4*)(agg + (size_t)cr * D + c4 * 4);
        float4 vb = *(const float4*)(e_r_bias + (size_t)cr * D + c4 * 4);
        float* da = &sAgg[r * LDSTRIDE + c4 * 4];
        float* db = &sBias[r * LDSTRIDE + c4 * 4];
        da[0] = va.x * on; da[1] = va.y * on; da[2] = va.z * on; da[3] = va.w * on;
        db[0] = vb.x; db[1] = vb.y; db[2] = vb.z; db[3] = vb.w;
    }
    __syncthreads();

    const int nc = lane & 15, roff = (lane >> 4) * 8;

    for (int jt = 0; jt < 8; ++jt) {        // 16-col chunk of each gate
        const int j = jt * 16 + nc;         // gate-local column 0..127
        v8f cr0, cr1, cz0, cz1, cxn0, cxn1, chn0, chn1;
        {
            float vr = b_ih[j] + b_hh[j];
            float vz = b_ih[128 + j] + b_hh[128 + j];
            float vx = b_ih[256 + j];
            float vh = b_hh[256 + j];
#pragma unroll
            for (int v = 0; v < 8; ++v) {
                cr0[v] = vr; cr1[v] = vr; cz0[v] = vz; cz1[v] = vz;
                cxn0[v] = vx; cxn1[v] = vx; chn0[v] = vh; chn1[v] = vh;
            }
        }
        // agg_scaled @ w_ih^T[0:128,:]  (K = 0..127)
#pragma unroll 4
        for (int ks = 0; ks < 32; ++ks) {
            const int k = ks * 4;
            v2f a0 = loadA(sAgg, 0, k, lane);
            v2f a1 = loadA(sAgg, 16, k, lane);
            v2f br = loadBP(ihp, 64, jt, ks, lane);
            v2f bz = loadBP(ihp, 64, 8 + jt, ks, lane);
            v2f bn = loadBP(ihp, 64, 16 + jt, ks, lane);
            cr0 = wmma4(a0, br, cr0);  cr1 = wmma4(a1, br, cr1);
            cz0 = wmma4(a0, bz, cz0);  cz1 = wmma4(a1, bz, cz1);
            cxn0 = wmma4(a0, bn, cxn0); cxn1 = wmma4(a1, bn, cxn1);
        }
        // e_r_bias @ w_ih^T[128:256,:]  (K = 128..255)
#pragma unroll 4
        for (int ks = 0; ks < 32; ++ks) {
            const int k = ks * 4;
            v2f a0 = loadA(sBias, 0, k, lane);
            v2f a1 = loadA(sBias, 16, k, lane);
            v2f br = loadBP(ihp, 64, jt, 32 + ks, lane);
            v2f bz = loadBP(ihp, 64, 8 + jt, 32 + ks, lane);
            v2f bn = loadBP(ihp, 64, 16 + jt, 32 + ks, lane);
            cr0 = wmma4(a0, br, cr0);  cr1 = wmma4(a1, br, cr1);
            cz0 = wmma4(a0, bz, cz0);  cz1 = wmma4(a1, bz, cz1);
            cxn0 = wmma4(a0, bn, cxn0); cxn1 = wmma4(a1, bn, cxn1);
        }
        // e_r_bias @ w_hh^T  (K = 0..127)
#pragma unroll 4
        for (int ks = 0; ks < 32; ++ks) {
            const int k = ks * 4;
            v2f a0 = loadA(sBias, 0, k, lane);
            v2f a1 = loadA(sBias, 16, k, lane);
            v2f br = loadBP(hhp, 32, jt, ks, lane);
            v2f bz = loadBP(hhp, 32, 8 + jt, ks, lane);
            v2f bn = loadBP(hhp, 32, 16 + jt, ks, lane);
            cr0 = wmma4(a0, br, cr0);  cr1 = wmma4(a1, br, cr1);
            cz0 = wmma4(a0, bz, cz0);  cz1 = wmma4(a1, bz, cz1);
            chn0 = wmma4(a0, bn, chn0); chn1 = wmma4(a1, bn, chn1);
        }
        // gates + relu, write pre-normalized h0
#pragma unroll
        for (int half = 0; half < 2; ++half) {
#pragma unroll
            for (int v = 0; v < 8; ++v) {
                float xr = half ? cr1[v] : cr0[v];
                float xz = half ? cz1[v] : cz0[v];
                float xn = half ? cxn1[v] : cxn0[v];
                float hn = half ? chn1[v] : chn0[v];
                float r = 1.0f / (1.0f + __expf(-xr));
                float z = 1.0f / (1.0f + __expf(-xz));
                float nn = tanhf(xn + r * hn);
                int lr = half * 16 + v + roff;
                float hprev = sBias[lr * LDSTRIDE + jt * 16 + nc];
                float h = (1.0f - z) * nn + z * hprev;
                h = fmaxf(h, 0.0f);
                int gr = rowbase + lr;
                if (full) {
                    out[(size_t)gr * D + jt * 16 + nc] = h;
                } else if (gr < N) {
                    out[(size_t)gr * D + jt * 16 + nc] = h;
                }
            }
        }
    }
}

// ---------------- K4: row L2 normalize (1 wave / row) --------------------------
__global__ __launch_bounds__(256) void k_norm(float* __restrict__ out, int N) {
    int idx = blockIdx.x * 256 + threadIdx.x;
    int row = idx >> 5, lane = idx & 31;
    if (row >= N) return;
    float4 v = *(float4*)(out + (size_t)row * D + lane * 4);
    float s = v.x * v.x + v.y * v.y + v.z * v.z + v.w * v.w;
#pragma unroll
    for (int off = 16; off > 0; off >>= 1) s += __shfl_xor(s, off, 32);
    float inv = 1.0f / fmaxf(sqrtf(s), 1e-12f);
    v.x *= inv; v.y *= inv; v.z *= inv; v.w *= inv;
    *(float4*)(out + (size_t)row * D + lane * 4) = v;
}

extern "C" void kernel_launch(void* const* d_in, const int* in_sizes, int n_in,
                              void* d_out, int out_size, void* d_ws, size_t ws_size,
                              hipStream_t stream) {
    const float* ent      = (const float*)d_in[0];
    const float* e_r_bias = (const float*)d_in[2];
    const float* out_norm = (const float*)d_in[3];
    const float* W        = (const float*)d_in[4];   // [128,128]
    const float* w_ih     = (const float*)d_in[5];   // [384,256]
    const float* w_hh     = (const float*)d_in[6];   // [384,128]
    const float* b_ih     = (const float*)d_in[7];
    const float* b_hh     = (const float*)d_in[8];
    const int*   node_id  = (const int*)d_in[9];
    const int*   esrc     = (const int*)d_in[10];
    const int*   edst     = (const int*)d_in[11];

    const int N = in_sizes[3];        // out_norm has N elements
    const int E = in_sizes[10];
    float* out = (float*)d_out;

    float* hW  = (float*)d_ws;                      // N*128
    float* agg = hW + (size_t)N * D;                // N*128
    float* Wp  = agg + (size_t)N * D;               // 128*128 packed
    float* ihp = Wp + 128 * 128;                    // 256*384 packed
    float* hhp = ihp + 256 * 384;                   // 128*384 packed

    hipMemsetAsync(agg, 0, (size_t)N * D * sizeof(float), stream);

    // pack weights into WMMA fragment-major layout (absorbs transposes)
    k_pack_direct<<<(8 * 32 * 32 + 255) / 256, 256, 0, stream>>>(W, Wp, 32, 8, 128);
    k_pack_transposed<<<(24 * 64 * 32 + 255) / 256, 256, 0, stream>>>(w_ih, ihp, 64, 24, 256);
    k_pack_transposed<<<(24 * 32 * 32 + 255) / 256, 256, 0, stream>>>(w_hh, hhp, 32, 24, 128);

    k_gather_gemm<<<(N + 63) / 64, 64, 0, stream>>>(ent, node_id, (const v2f*)Wp, hW, N);

    long long sthreads = (long long)E * 32;
    k_scatter<<<(unsigned)((sthreads + 255) / 256), 256, 0, stream>>>(hW, esrc, edst, agg, E);

    k_gru<<<(N + 31) / 32, 32, 0, stream>>>(agg, out_norm, e_r_bias,
                                            (const v2f*)ihp, (const v2f*)hhp,
                                            b_ih, b_hh, out, N);

    long long nthreads = (long long)N * 32;
    k_norm<<<(unsigned)((nthreads + 255) / 256), 256, 0, stream>>>(out, N);
}